// CustomEmbedding_19335942767147
// MI455X (gfx1250) — compile-verified
//
#include <hip/hip_runtime.h>
#include <hip/hip_bf16.h>

// Embedding gather: out[row, :] = weight[x[row], :]
//   rows = B*S = 4096, D = 512 f32  ->  128 float4 per row.
// Pure bandwidth problem (~16.8 MB total traffic, ~0.7us at 23.3 TB/s).
// One row is moved by 128 consecutive threads; each thread moves one 16B
// packet (fully coalesced: 512B per wave32 memory op).
//
// CDNA5 path: stage each 16B packet through LDS with the gfx1250 async copy
// ops GLOBAL_LOAD_ASYNC_TO_LDS_B128 / GLOBAL_STORE_ASYNC_FROM_LDS_B128
// (ASYNCcnt-tracked, VGPR-data-free DMA-style copy), synced with
// s_wait_asynccnt. Falls back to a plain b128 gather if builtins are absent.

#define GAS __attribute__((address_space(1)))
#define LAS __attribute__((address_space(3)))

typedef int v4i __attribute__((ext_vector_type(4)));

#if defined(__HIP_DEVICE_COMPILE__) &&                                   \
    __has_builtin(__builtin_amdgcn_global_load_async_to_lds_b128) &&     \
    __has_builtin(__builtin_amdgcn_global_store_async_from_lds_b128)
#define EMB_USE_ASYNC_LDS 1
#else
#define EMB_USE_ASYNC_LDS 0
#endif

__global__ __launch_bounds__(256) void emb_gather_kernel(
    const int* __restrict__ idx,      // [n_rows] token ids
    const float4* __restrict__ w4,    // [V, 128] weight rows as float4
    float4* __restrict__ out4,        // [n_rows, 128]
    int n_rows)
{
    const int tid = blockIdx.x * 256 + threadIdx.x;
    const int row = tid >> 7;          // 128 threads per row
    const int col = tid & 127;
    if (row >= n_rows) return;

    // row is wave-uniform (groups of 128 threads fully contain each wave32);
    // readfirstlane lets the compiler keep the index load scalar.
    const int row_u = __builtin_amdgcn_readfirstlane(row);
    const int e = idx[row_u];

    const float4* src = w4 + (size_t)e * 128 + col;
    float4* dst = out4 + (size_t)row * 128 + col;

#if EMB_USE_ASYNC_LDS
    __shared__ float4 stage[256];               // 4 KB per block
    LAS v4i* lp = (LAS v4i*)&stage[threadIdx.x];
    // global -> LDS (async, per-lane b128), wait ASYNCcnt, LDS -> global.
    __builtin_amdgcn_global_load_async_to_lds_b128((GAS v4i*)src, lp, 0, 0);
#if __has_builtin(__builtin_amdgcn_s_wait_asynccnt)
    __builtin_amdgcn_s_wait_asynccnt(0);
#else
    asm volatile("s_wait_asynccnt 0" ::: "memory");
#endif
    __builtin_amdgcn_global_store_async_from_lds_b128((GAS v4i*)dst, lp, 0, 0);
    // S_ENDPGM performs an implicit wait-idle, so the async store drains
    // before the wave retires.
#else
    *dst = *src;                                 // b128 load + b128 store
#endif
}

extern "C" void kernel_launch(void* const* d_in, const int* in_sizes, int n_in,
                              void* d_out, int out_size, void* d_ws, size_t ws_size,
                              hipStream_t stream) {
    const int*   x = (const int*)d_in[0];      // [B*S] token ids
    const float* w = (const float*)d_in[1];    // [V, 512]
    float*     out = (float*)d_out;            // [B*S, 512]

    const int n_rows = in_sizes[0];            // B*S = 4096
    const int total_f4 = n_rows * 128;         // one float4 per thread
    const int block = 256;
    const int grid = (total_f4 + block - 1) / block;

    emb_gather_kernel<<<grid, block, 0, stream>>>(
        x, (const float4*)w, (float4*)out, n_rows);
}